// ObsDMLLoss_30743375904834
// MI455X (gfx1250) — compile-verified
//
#include <hip/hip_runtime.h>
#include <math.h>

// ObsDMLLoss for MI455X (gfx1250, wave32).
// Kernel 1: streaming, memory/trans-roofline-bound DML math, 4 items/thread,
//           deterministic LDS tree -> 1 partial per block (1024 partials).
// Kernel 2: single wave32; float4 dual-stream accumulation (latency-hidden),
//           then V_WMMA_F32_16X16X4_F32 exact-f32 cross-lane reduce.

typedef __attribute__((ext_vector_type(2))) float v2f;
typedef __attribute__((ext_vector_type(8))) float v8f;

#define NR_MIX 10
#define IPT 4                 // items per thread in kernel 1
#define BLK 256

__device__ __forceinline__ float dml_item(
    const float* __restrict__ x, const float* __restrict__ l,
    int i, float nb, float half, float log_halfbins)
{
    const float xv = x[i];

    // 30 contiguous floats at byte offset i*120 (8B aligned) -> float2 loads.
    float v[3 * NR_MIX];
    const float2* p = (const float2*)(l + (size_t)i * (3 * NR_MIX));
    #pragma unroll
    for (int j = 0; j < (3 * NR_MIX) / 2; ++j) {
        float2 t2 = p[j];
        v[2 * j]     = t2.x;
        v[2 * j + 1] = t2.y;
    }

    // log_softmax over logits v[0..9]
    float mx = v[0];
    #pragma unroll
    for (int k = 1; k < NR_MIX; ++k) mx = fmaxf(mx, v[k]);
    float sume = 0.0f;
    #pragma unroll
    for (int k = 0; k < NR_MIX; ++k) sume += expf(v[k] - mx);
    const float lse = logf(sume);

    float tk[NR_MIX];
    #pragma unroll
    for (int k = 0; k < NR_MIX; ++k) {
        const float mean = v[NR_MIX + k];
        const float lsc  = fmaxf(v[2 * NR_MIX + k], -7.0f);
        const float inv  = expf(-lsc);
        const float cen  = xv - mean;
        const float pin  = inv * (cen + half);
        const float min_ = inv * (cen - half);
        const float mid  = inv * cen;

        // Shared exp(-|z|) for stable sigmoid & softplus.
        const float tp = expf(-fabsf(pin));
        const float tm = expf(-fabsf(min_));
        const float td = expf(-fabsf(mid));

        const float sp_p   = fmaxf(pin,  0.0f) + log1pf(tp);
        const float sp_m   = fmaxf(min_, 0.0f) + log1pf(tm);
        const float sp_mid = fmaxf(mid,  0.0f) + log1pf(td);

        const float sig_p = (pin  >= 0.0f) ? (1.0f / (1.0f + tp)) : (tp / (1.0f + tp));
        const float sig_m = (min_ >= 0.0f) ? (1.0f / (1.0f + tm)) : (tm / (1.0f + tm));
        const float cdf_delta = sig_p - sig_m;

        const float log_cdf_plus = pin - sp_p;
        const float log_1m_cdf   = -sp_m;
        const float log_pdf_mid  = mid - lsc - 2.0f * sp_mid;

        float inner = (cdf_delta > 1e-7f)
                          ? logf(fmaxf(cdf_delta, 1e-12f))
                          : (log_pdf_mid - log_halfbins);
        float lp = (xv < -0.9999f) ? log_cdf_plus
                 : ((xv > 0.9999f) ? log_1m_cdf : inner);
        lp = (nb > 1.5f) ? lp : 0.0f;

        tk[k] = lp + (v[k] - mx - lse);
    }

    // logsumexp over mixtures
    float M = tk[0];
    #pragma unroll
    for (int k = 1; k < NR_MIX; ++k) M = fmaxf(M, tk[k]);
    float s = 0.0f;
    #pragma unroll
    for (int k = 0; k < NR_MIX; ++k) s += expf(tk[k] - M);
    return M + logf(s);
}

__global__ __launch_bounds__(BLK) void dml_main_kernel(
    const float* __restrict__ x,
    const float* __restrict__ l,
    const int*   __restrict__ bins,
    float*       __restrict__ partials,
    int n, int D)
{
    const int tid  = threadIdx.x;
    const int base = blockIdx.x * (BLK * IPT);

    float acc = 0.0f;
    const int i0 = base + tid;
    if (i0 < n) {
        // All IPT items of this thread share d: (i0 + 256*jj) % 64 == i0 % 64
        // (holds since BLK % D == 0 for D == 64; recomputed per item otherwise).
        const bool dconst = (BLK % D) == 0;
        int   d    = i0 % D;
        float nb   = (float)bins[d];
        float nbm1 = fmaxf(nb - 1.0f, 1.0f);
        float half = 1.0f / nbm1;
        float lhb  = logf(nbm1 * 0.5f);

        #pragma unroll
        for (int jj = 0; jj < IPT; ++jj) {
            const int i = i0 + jj * BLK;
            if (i < n) {
                if (!dconst && jj > 0) {
                    d    = i % D;
                    nb   = (float)bins[d];
                    nbm1 = fmaxf(nb - 1.0f, 1.0f);
                    half = 1.0f / nbm1;
                    lhb  = logf(nbm1 * 0.5f);
                }
                acc += dml_item(x, l, i, nb, half, lhb);
            }
        }
    }

    // Deterministic block tree reduction (fixed order every call).
    __shared__ float red[BLK];
    red[tid] = acc;
    __syncthreads();
    #pragma unroll
    for (int sft = BLK / 2; sft > 0; sft >>= 1) {
        if (tid < sft) red[tid] += red[tid + sft];
        __syncthreads();
    }
    if (tid == 0) partials[blockIdx.x] = red[0];
}

// Single-wave final reduction. Dual-stream float4 loads (2 outstanding b128
// loads per iteration) hide L2 latency; fixed accumulation order keeps the
// result bitwise deterministic. Then V_WMMA_F32_16X16X4_F32 with A = ones(16x4),
// B = the 64 lane partials (2 VGPRs x 32 lanes), C = 0: lane L's first D VGPR
// holds column-sum (L&15). Summing all 16 column sums equals the sum of all 64
// B slots under any valid slot->column partition, so the total is
// layout-robust. EXEC is all ones (one full wave, converged at the WMMA).
__global__ __launch_bounds__(32) void dml_reduce_kernel(
    const float* __restrict__ partials,
    float*       __restrict__ out,
    int nparts, float bsize)
{
    const int t = threadIdx.x;  // 0..31

    const float4* p4 = (const float4*)partials;  // d_ws is 16B-aligned
    const int n4 = nparts >> 2;

    float4 a0 = make_float4(0.f, 0.f, 0.f, 0.f);
    float4 a1 = make_float4(0.f, 0.f, 0.f, 0.f);
    for (int j = t; j < n4; j += 64) {
        float4 u = p4[j];
        a0.x += u.x; a0.y += u.y; a0.z += u.z; a0.w += u.w;
        const int j2 = j + 32;
        if (j2 < n4) {
            float4 w = p4[j2];
            a1.x += w.x; a1.y += w.y; a1.z += w.z; a1.w += w.w;
        }
    }
    // Scalar tail for nparts not divisible by 4.
    for (int j = (n4 << 2) + t; j < nparts; j += 32) a0.x += partials[j];

    const float s0 = (a0.x + a0.y) + (a0.z + a0.w);
    const float s1 = (a1.x + a1.y) + (a1.z + a1.w);

    v2f a; a[0] = 1.0f; a[1] = 1.0f;   // A = ones(16x4)
    v2f b; b[0] = s0;   b[1] = s1;     // B = 4x16 of lane partials
    v8f c = {};
    // (neg_a, A, neg_b, B, c_mod, C, reuse_a, reuse_b)
    v8f acc = __builtin_amdgcn_wmma_f32_16x16x4_f32(
        false, a, false, b, (short)0, c, false, false);

    __shared__ float red[16];
    if (t < 16) red[t] = acc[0];       // 16 distinct column sums
    __syncthreads();

    if (t == 0) {
        float tot = 0.0f;
        #pragma unroll
        for (int k = 0; k < 16; ++k) tot += red[k];
        const float neg = -tot;
        out[0] = neg;
        out[1] = neg / (bsize * 0.69314718f);  // np.float32(log(2))
    }
}

extern "C" void kernel_launch(void* const* d_in, const int* in_sizes, int n_in,
                              void* d_out, int out_size, void* d_ws, size_t ws_size,
                              hipStream_t stream) {
    const float* x    = (const float*)d_in[0];   // (B, D, 1) f32
    const float* l    = (const float*)d_in[1];   // (B, D, 30) f32
    const int*   bins = (const int*)  d_in[2];   // (D,) i32
    // d_in[3] (mask) is unused by the reference.

    const int n = in_sizes[0];      // B*D = 1,048,576
    const int D = in_sizes[2];      // 64
    const int nblk = (n + BLK * IPT - 1) / (BLK * IPT);  // 1024

    float* partials = (float*)d_ws; // nblk floats (4 KB) << ws_size

    dml_main_kernel<<<nblk, BLK, 0, stream>>>(x, l, bins, partials, n, D);
    dml_reduce_kernel<<<1, 32, 0, stream>>>(partials, (float*)d_out, nblk,
                                            (float)(n / D));
}